// EmbeddingBlock_50448685859051
// MI455X (gfx1250) — compile-verified
//
#include <hip/hip_runtime.h>
#include <hip/hip_bf16.h>
#include <stdint.h>

#define NSPH 7
#define NRAD 6
#define BLK  128           // threads per block == triplets per block
#define INV4PI 0.07957747154594767f
#define SQRT2F 1.41421356237309515f

// zeros z_{l,i} of spherical Bessel j_l, l=0..6, i=0..5
__device__ const float Z_TAB[7][6] = {
  { 3.14159265f,  6.28318531f,  9.42477796f, 12.56637061f, 15.70796327f, 18.84955592f},
  { 4.49340946f,  7.72525184f, 10.90412166f, 14.06619391f, 17.22075527f, 20.37130296f},
  { 5.76345920f,  9.09501133f, 12.32294097f, 15.51460301f, 18.68903636f, 21.85387408f},
  { 6.98793200f, 10.41711855f, 13.69802315f, 16.92362129f, 20.12180617f, 23.30424693f},
  { 8.18256145f, 11.70490715f, 15.03966471f, 18.30125595f, 21.52541771f, 24.72756581f},
  { 9.35581211f, 12.96653017f, 16.35470964f, 19.65315286f, 22.90455063f, 26.12775052f},
  {10.51283540f, 14.20739236f, 17.64797496f, 20.98346482f, 24.26276804f, 27.50786839f}
};

__device__ __forceinline__ float frcp(float x) { return __builtin_amdgcn_rcpf(x); }

// ---- CDNA5 async-tensor path: per-lane async copy global -> LDS (ASYNCcnt) ----
__device__ __forceinline__ void async_g2lds_b32(void* lds_dst, const void* gbase, unsigned byte_off) {
  unsigned lds = (unsigned)(uintptr_t)lds_dst;   // generic addr low 32 bits == LDS offset
  asm volatile("global_load_async_to_lds_b32 %0, %1, %2"
               :: "v"(lds), "v"(byte_off), "s"(gbase) : "memory");
}
__device__ __forceinline__ void wait_asynccnt0() {
  asm volatile("s_wait_asynccnt 0" ::: "memory");
}

// ---- kernel 1: dist_emb [E,6] : envelope(x) * sin(freq_r * x) ----
__global__ void __launch_bounds__(256) dist_emb_kernel(
    const float* __restrict__ dist, const float* __restrict__ freq,
    float* __restrict__ out, int E) {
  int g = blockIdx.x * blockDim.x + threadIdx.x;
  if (g >= E * NRAD) return;
  int e = g / NRAD;
  int r = g - e * NRAD;
  float x    = dist[e] * 0.2f;                     // dist / CUTOFF
  float invx = frcp(x);
  float x2   = x * x;
  float x5   = x2 * x2 * x;
  float env  = invx + x5 * (-28.0f + 48.0f * x - 21.0f * x2);
  out[g] = env * __sinf(freq[r] * x);
}

// ---- kernel 2: angle_emb [T,42] + torsion_emb [T,294] ----
__global__ void __launch_bounds__(BLK) triplet_kernel(
    const float* __restrict__ dist, const float* __restrict__ angle,
    const float* __restrict__ torsion, const int* __restrict__ idx_kj,
    float* __restrict__ out_ang, float* __restrict__ out_tor, int T, int E) {

  __shared__ float sh_rbf [BLK * 42];
  __shared__ float sh_cbfT[BLK * 49];
  __shared__ float sh_cbfA[BLK * 7];
  __shared__ float sh_norm[42];
  __shared__ float sh_K2  [49];
  __shared__ float sh_ang [BLK];
  __shared__ float sh_tor [BLK];

  const int tid = threadIdx.x;
  const int t0  = blockIdx.x * BLK;
  const int tt  = min(t0 + tid, T - 1);            // clamped for tail block

  // stage angle/torsion tiles via CDNA5 async copy (overlaps with phase 0 below)
  async_g2lds_b32(&sh_ang[tid], angle,   (unsigned)tt * 4u);
  async_g2lds_b32(&sh_tor[tid], torsion, (unsigned)tt * 4u);
  const int gidx = idx_kj[tt];
  __builtin_prefetch(&dist[gidx], 0, 0);           // global_prefetch for the L2-resident gather

  // ---- phase 0: per-block constant tables ----
  if (tid < 49) {                                  // spherical-harmonic normalizers K
    int c = tid, l = 0;
    while ((l + 1) * (l + 1) <= c) ++l;
    int m  = c - l * l - l;
    int am = m < 0 ? -m : m;
    float prod = 1.0f;
    for (int q = l - am + 1; q <= l + am; ++q) prod *= (float)q;
    float K = sqrtf((float)(2 * l + 1) * INV4PI / prod);
    if (m != 0) K *= SQRT2F;
    sh_K2[c] = K;
  }
  if (tid >= 64 && tid < 64 + 42) {                // Bessel normalizers sqrt(2)/|j_{l+1}(z)|
    int li = tid - 64, l = li / 6, i = li - l * 6;
    float x = Z_TAB[l][i];
    float s = sinf(x), cc = cosf(x);
    float invx = frcp(x);
    float jm = s * invx;
    float j  = (s * invx - cc) * invx;             // j1
    for (int ll = 2; ll <= l + 1; ++ll) { float jn = (float)(2 * ll - 1) * invx * j - jm; jm = j; j = jn; }
    sh_norm[li] = SQRT2F * frcp(fabsf(j));
  }

  wait_asynccnt0();
  __syncthreads();

  // ---- phase 1: one thread computes one triplet's basis values ----
  {
    const float d = dist[gidx] * 0.2f;             // dist_scaled in (0.1, 1]
    const float invd = frcp(d);
    // rbf rows: norm * j_l(z * d)
#pragma unroll
    for (int l = 0; l < NSPH; ++l) {
#pragma unroll
      for (int i = 0; i < NRAD; ++i) {
        float x = d * Z_TAB[l][i];
        float s, cc; __sincosf(x, &s, &cc);
        float invx = invd * (1.0f / Z_TAB[l][i]);  // 1/(d*z): rcp(d) * const reciprocal
        float j0 = s * invx, j;
        if (l == 0) j = j0;
        else {
          float jm = j0;
          j = (s * invx - cc) * invx;
#pragma unroll
          for (int ll = 2; ll <= l; ++ll) { float jn = (float)(2 * ll - 1) * invx * j - jm; jm = j; j = jn; }
        }
        sh_rbf[tid * 42 + l * 6 + i] = sh_norm[l * 6 + i] * j;
      }
    }
    // associated Legendre table (Condon-Shortley), fully in registers
    const float ang = sh_ang[tid], phi = sh_tor[tid];
    float ct, st; __sincosf(ang, &st, &ct);
    float P[7][7];
    P[0][0] = 1.0f;
#pragma unroll
    for (int m = 1; m <= 6; ++m) P[m][m] = (float)(1 - 2 * m) * P[m - 1][m - 1] * st;
#pragma unroll
    for (int m = 0; m <= 5; ++m) P[m + 1][m] = (float)(2 * m + 1) * ct * P[m][m];
#pragma unroll
    for (int m = 0; m <= 6; ++m)
#pragma unroll
      for (int l = m + 2; l <= 6; ++l)
        P[l][m] = ((float)(2 * l - 1) * ct * P[l - 1][m] - (float)(l + m - 1) * P[l - 2][m])
                  * (1.0f / (float)(l - m));        // compile-time reciprocal, no div chain
    // cbf_angle (m=0 harmonics)
#pragma unroll
    for (int l = 0; l <= 6; ++l)
      sh_cbfA[tid * 7 + l] = sqrtf((float)(2 * l + 1) * INV4PI) * P[l][0];
    // cbf_torsion (full real harmonics) via sin/cos(m*phi) recurrence
    float sm[7], cm[7];
    sm[0] = 0.0f; cm[0] = 1.0f;
    __sincosf(phi, &sm[1], &cm[1]);
#pragma unroll
    for (int m = 2; m <= 6; ++m) {
      sm[m] = sm[m - 1] * cm[1] + cm[m - 1] * sm[1];
      cm[m] = cm[m - 1] * cm[1] - sm[m - 1] * sm[1];
    }
    sh_cbfT[tid * 49 + 0] = 0.28209479177387814f;  // 0.5/sqrt(pi)
#pragma unroll
    for (int l = 1; l <= 6; ++l)
#pragma unroll
      for (int m = -l; m <= l; ++m) {
        int am = m < 0 ? -m : m;
        float trig = (m < 0) ? sm[am] : cm[am];    // cm[0]==1 handles m==0
        sh_cbfT[tid * 49 + l * l + l + m] = sh_K2[l * l + l + m] * trig * P[l][am];
      }
  }
  __syncthreads();

  // ---- phase 2: coalesced stores; thread owns fixed column e, loops over triplets ----
  const int nlt = min(BLK, T - t0);
  auto emit = [&](int e) {
    if (e < 294) {                                  // torsion: out[t, i, j, k] = rbf[j*6+k]*cbfT[i*7+j]
      int a1 = e % 42;
      int a2 = (e / 42) * 7 + a1 / 6;
      const float* rb = sh_rbf + a1;
      const float* cb = sh_cbfT + a2;
      float* dst = out_tor + (size_t)t0 * 294 + e;
#pragma unroll 4
      for (int lt = 0; lt < nlt; ++lt)
        dst[(size_t)lt * 294] = rb[lt * 42] * cb[lt * 49];
    } else {                                        // angle: out[t, s] = rbf[s]*cbfA[s/6]
      int s = e - 294;
      const float* rb = sh_rbf + s;
      const float* cb = sh_cbfA + s / 6;
      float* dst = out_ang + (size_t)t0 * 42 + s;
#pragma unroll 4
      for (int lt = 0; lt < nlt; ++lt)
        dst[(size_t)lt * 42] = rb[lt * 42] * cb[lt * 7];
    }
  };
  emit(tid);                 // columns   0..127  (torsion)
  emit(tid + BLK);           // columns 128..255  (torsion)
  if (tid < 80) emit(tid + 2 * BLK);   // columns 256..335 (torsion tail + all 42 angle cols)
}

extern "C" void kernel_launch(void* const* d_in, const int* in_sizes, int n_in,
                              void* d_out, int out_size, void* d_ws, size_t ws_size,
                              hipStream_t stream) {
  (void)n_in; (void)out_size; (void)d_ws; (void)ws_size;
  const float* dist    = (const float*)d_in[0];
  const float* angle   = (const float*)d_in[1];
  const float* torsion = (const float*)d_in[2];
  const float* freq    = (const float*)d_in[3];
  const int*   idx_kj  = (const int*)d_in[4];
  const int E = in_sizes[0];
  const int T = in_sizes[1];

  float* out      = (float*)d_out;
  float* out_ang  = out + (size_t)E * NRAD;                       // after dist_emb [E,6]
  float* out_tor  = out_ang + (size_t)T * (NSPH * NRAD);          // after angle_emb [T,42]

  int n_dist = E * NRAD;
  dist_emb_kernel<<<(n_dist + 255) / 256, 256, 0, stream>>>(dist, freq, out, E);

  int n_blocks = (T + BLK - 1) / BLK;
  triplet_kernel<<<n_blocks, BLK, 0, stream>>>(dist, angle, torsion, idx_kj,
                                               out_ang, out_tor, T, E);
}